// DenseDilatedKnnGraph_6296422056176
// MI455X (gfx1250) — compile-verified
//
#include <hip/hip_runtime.h>
#include <hip/hip_bf16.h>
#include <float.h>

typedef __attribute__((ext_vector_type(2))) float v2f;
typedef __attribute__((ext_vector_type(8))) float v8f;

#define NB   4
#define DIM  64
#define NPTS 8192
#define KK   9
#define LB   90

// Kernel 1: L2-normalize each point's 64-channel vector; write transposed
// xt[b][n][d] (contiguous per point) and sq[b][n] = ||x||^2 after normalize.
__global__ __launch_bounds__(256)
void DenseKnn_normalize(const float* __restrict__ x,
                        float* __restrict__ xt,
                        float* __restrict__ sq) {
    int g = blockIdx.x * blockDim.x + threadIdx.x;   // 0 .. NB*NPTS-1
    int b = g >> 13;                                 // / 8192
    int n = g & (NPTS - 1);
    const float* xp = x + (size_t)b * DIM * NPTS + n;
    float v[DIM];
    float s = 0.0f;
#pragma unroll
    for (int d = 0; d < DIM; ++d) {
        v[d] = xp[(size_t)d * NPTS];
        s += v[d] * v[d];
    }
    float inv = 1.0f / fmaxf(sqrtf(s), 1e-12f);
    float* o = xt + ((size_t)b * NPTS + n) * DIM;
    float s2 = 0.0f;
#pragma unroll
    for (int d = 0; d < DIM; ++d) {
        float t = v[d] * inv;
        o[d] = t;
        s2 += t * t;
    }
    sq[(size_t)b * NPTS + n] = s2;
}

// Kernel 2: one wave32 per 16-row tile. Banded 16x112 dot-product block via
// V_WMMA_F32_16X16X4_F32 (K=64 in 16 steps), scores -> LDS, then per-row
// stable top-9 insertion selection + reference head fix.
__global__ __launch_bounds__(128)
void DenseKnn_knn(const float* __restrict__ xt,
                  const float* __restrict__ sq,
                  int* __restrict__ out) {
    __shared__ float scores[4][16][112];   // [wave][row][col], 28 KB

    const int lane = threadIdx.x & 31;
    const int wave = threadIdx.x >> 5;
    const int tilesPerBatch = NPTS / 64;   // 128
    const int b    = blockIdx.x / tilesPerBatch;
    const int tile = blockIdx.x % tilesPerBatch;
    const int r0   = tile * 64 + wave * 16;          // tile rows [r0, r0+16)

    const float* xb  = xt + (size_t)b * NPTS * DIM;
    const float* sqb = sq + (size_t)b * NPTS;

    const int lrow = lane & 15;            // M (for A) / N (for B) index
    const int koff = (lane >> 4) * 2;      // K sub-offset within a step of 4

    // ---- A fragment: rows r0..r0+15, f32 16x4 layout per step s ----
    v2f a[16];
    {
        const float* ap = xb + (size_t)(r0 + lrow) * DIM + koff;
#pragma unroll
        for (int s = 0; s < 16; ++s)
            a[s] = *(const v2f*)(ap + 4 * s);
    }

    // ---- 7 column tiles covering [r0-96, r0+16) ----
#pragma unroll
    for (int ct = 0; ct < 7; ++ct) {
        const int c0  = r0 - 96 + ct * 16;
        int col  = c0 + lrow;
        int colc = col < 0 ? 0 : col;      // col <= r0+15 <= N-1 always
        v2f bv[16];
        const float* bp = xb + (size_t)colc * DIM + koff;
#pragma unroll
        for (int s = 0; s < 16; ++s)
            bv[s] = *(const v2f*)(bp + 4 * s);

        v8f acc = {};
#pragma unroll
        for (int s = 0; s < 16; ++s)
            acc = __builtin_amdgcn_wmma_f32_16x16x4_f32(
                      /*neg_a=*/false, a[s], /*neg_b=*/false, bv[s],
                      /*c_mod=*/(short)0, acc, /*reuse_a=*/false, /*reuse_b=*/false);

        // score = sq_m - 2*dot   (sq_n is a per-row constant: order-invariant)
        const float sqc = sqb[colc];
        const int rowHalf = (lane >> 4) * 8;   // D layout: VGPR j -> row j or j+8
#pragma unroll
        for (int j = 0; j < 8; ++j)
            scores[wave][rowHalf + j][ct * 16 + lrow] = sqc - 2.0f * acc[j];
    }

    __syncthreads();

    // ---- per-row top-9 selection (lanes 0..15 each own one row) ----
    if (lane < 16) {
        const int r = r0 + lane;
        float bs[KK]; int bi[KK];
#pragma unroll
        for (int j = 0; j < KK; ++j) { bs[j] = FLT_MAX; bi[j] = 0; }

        int mstart = r - (LB - 1);
        if (mstart < 0) mstart = 0;
        for (int m = mstart; m <= r; ++m) {
            float sc = scores[wave][lane][m - (r0 - 96)];
            if (sc < bs[KK - 1]) {          // strict <: ties keep lower index
                int j = KK - 1;
                while (j > 0 && sc < bs[j - 1]) {
                    bs[j] = bs[j - 1]; bi[j] = bi[j - 1]; --j;
                }
                bs[j] = sc; bi[j] = m;
            }
        }

        const size_t base0 = (((size_t)0 * NB + b) * NPTS + r) * KK;
        const size_t base1 = (((size_t)1 * NB + b) * NPTS + r) * KK;
#pragma unroll
        for (int j = 0; j < KK; ++j) {
            int v = (r < LB && j > r) ? bi[0] : bi[j];   // head fix (rows < 8)
            out[base0 + j] = v;
            out[base1 + j] = r;
        }
    }
}

extern "C" void kernel_launch(void* const* d_in, const int* in_sizes, int n_in,
                              void* d_out, int out_size, void* d_ws, size_t ws_size,
                              hipStream_t stream) {
    const float* x = (const float*)d_in[0];
    int* out = (int*)d_out;                 // output is int32 indices

    float* xt = (float*)d_ws;                               // 4*8192*64 f32 = 8 MB
    float* sq = xt + (size_t)NB * NPTS * DIM;               // 4*8192 f32 = 128 KB

    DenseKnn_normalize<<<(NB * NPTS) / 256, 256, 0, stream>>>(x, xt, sq);
    DenseKnn_knn<<<NB * (NPTS / 64), 128, 0, stream>>>(xt, sq, out);
}